// EmbeddedDropout_16973710754355
// MI455X (gfx1250) — compile-verified
//
#include <hip/hip_runtime.h>
#include <stdint.h>

// EmbeddedDropout gather: out[t,:] = weight[words[t],:] * ((mask_u[row] < 0.9) ? 1/0.9 : 0)
// Pure bandwidth kernel (~128 MiB moved -> ~6-8 us at 23.3 TB/s).
// CDNA5 path: double-buffered GLOBAL_LOAD_ASYNC_TO_LDS_B128 staging per wave,
// synchronized with S_WAIT_ASYNCCNT; non-temporal B128 stores for the
// write-once output so weight rows stay resident in the 192 MB L2.

typedef float v4f __attribute__((ext_vector_type(4)));
typedef int   v4i __attribute__((vector_size(16)));   // matches builtin param type

#define THREADS 256
#define ROW_D   1024                 // floats per embedding row (fast path)
#define KEEP_P  0.9f
#define INV_KEEP (1.0f / 0.9f)

// ---- CDNA5 async global->LDS helpers (guarded for the host pass) ----------
__device__ __forceinline__ void async_copy_b128(const float* g, float* l) {
#if defined(__gfx1250__)
#if __has_builtin(__builtin_amdgcn_global_load_async_to_lds_b128)
  __builtin_amdgcn_global_load_async_to_lds_b128(
      (__attribute__((address_space(1))) v4i*)(void*)g,
      (__attribute__((address_space(3))) v4i*)(void*)l,
      0, 0);
#else
  unsigned lds_off =
      (unsigned)(__SIZE_TYPE__)(__attribute__((address_space(3))) void*)l;
  asm volatile("global_load_async_to_lds_b128 %0, %1, off"
               :
               : "v"(lds_off), "v"(g)
               : "memory");
#endif
#else
  *(v4f*)l = *(const v4f*)g;  // non-gfx1250 / host pass placeholder
#endif
}

__device__ __forceinline__ void wait_async_le1() {
#if defined(__gfx1250__)
#if __has_builtin(__builtin_amdgcn_s_wait_asynccnt)
  __builtin_amdgcn_s_wait_asynccnt(1);
  asm volatile("" ::: "memory");
#else
  asm volatile("s_wait_asynccnt 1" ::: "memory");
#endif
#endif
}

__device__ __forceinline__ void wait_async_le0() {
#if defined(__gfx1250__)
#if __has_builtin(__builtin_amdgcn_s_wait_asynccnt)
  __builtin_amdgcn_s_wait_asynccnt(0);
  asm volatile("" ::: "memory");
#else
  asm volatile("s_wait_asynccnt 0" ::: "memory");
#endif
#endif
}

// ---- fast path: D == 1024, one token row per block iteration --------------
__global__ __launch_bounds__(THREADS)
void embdrop_async_kernel(const float* __restrict__ weight,
                          const float* __restrict__ mask_u,
                          const int*   __restrict__ words,
                          float* __restrict__ out,
                          int n_tokens) {
  __shared__ float smem[2 * ROW_D];   // two 4 KB row buffers

  const int tid  = threadIdx.x;
  const int foff = tid * 4;           // this thread's 16-byte slice of the row
  const int stride = gridDim.x;

  int   t   = (int)blockIdx.x;
  int   buf = 0;
  float scale_cur = 0.0f;

  // Prologue: stage the first token's row into buffer 0.
  if (t < n_tokens) {
    const int   row = words[t];                    // uniform broadcast load
    const float u   = mask_u[row];                 // uniform broadcast load
    scale_cur = (u < KEEP_P) ? INV_KEEP : 0.0f;
    async_copy_b128(weight + (size_t)row * ROW_D + foff, smem + foff);
  }

  while (t < n_tokens) {
    const int  tn        = t + stride;
    const bool have_next = (tn < n_tokens);
    float scale_next = 0.0f;

    if (have_next) {
      const int   rown = words[tn];
      const float un   = mask_u[rown];
      scale_next = (un < KEEP_P) ? INV_KEEP : 0.0f;
      async_copy_b128(weight + (size_t)rown * ROW_D + foff,
                      smem + (buf ^ 1) * ROW_D + foff);
      wait_async_le1();   // next is in flight; current buffer is complete
    } else {
      wait_async_le0();   // drain: current buffer complete
    }

    // Each lane consumes exactly the 16 LDS bytes it staged itself -> no barrier.
    v4f v = *(const v4f*)(smem + buf * ROW_D + foff);
    v *= scale_cur;
    __builtin_nontemporal_store(v, (v4f*)(out + (size_t)t * ROW_D + foff));

    buf ^= 1;
    scale_cur = scale_next;
    t = tn;
  }
}

// ---- generic fallback (any D), plain gather ------------------------------
__global__ void embdrop_generic_kernel(const float* __restrict__ weight,
                                       const float* __restrict__ mask_u,
                                       const int*   __restrict__ words,
                                       float* __restrict__ out,
                                       int n_tokens, int D) {
  long long i     = (long long)blockIdx.x * blockDim.x + threadIdx.x;
  long long total = (long long)n_tokens * D;
  long long step  = (long long)gridDim.x * blockDim.x;
  for (; i < total; i += step) {
    long long t = i / D;
    int       c = (int)(i - t * D);
    int   row = words[t];
    float u   = mask_u[row];
    float s   = (u < KEEP_P) ? INV_KEEP : 0.0f;
    out[i] = s * weight[(size_t)row * D + c];
  }
}

extern "C" void kernel_launch(void* const* d_in, const int* in_sizes, int n_in,
                              void* d_out, int out_size, void* d_ws, size_t ws_size,
                              hipStream_t stream) {
  (void)n_in; (void)d_ws; (void)ws_size;
  const float* weight = (const float*)d_in[0];   // (V, D) f32
  const float* mask_u = (const float*)d_in[1];   // (V, 1) f32
  const int*   words  = (const int*)d_in[2];     // (B, S) int32
  float*       out    = (float*)d_out;           // (B, S, D) f32

  const int n_tokens = in_sizes[2];
  const int D        = (n_tokens > 0) ? (out_size / n_tokens) : 0;

  if (D == ROW_D) {
    int grid = n_tokens < 4096 ? n_tokens : 4096;   // persistent; >=2 tokens/block for overlap
    if (grid < 1) grid = 1;
    embdrop_async_kernel<<<dim3(grid), dim3(THREADS), 0, stream>>>(
        weight, mask_u, words, out, n_tokens);
  } else {
    long long total  = (long long)n_tokens * D;
    int blocks = (int)((total + THREADS - 1) / THREADS);
    if (blocks > 65535) blocks = 65535;
    if (blocks < 1) blocks = 1;
    embdrop_generic_kernel<<<dim3(blocks), dim3(THREADS), 0, stream>>>(
        weight, mask_u, words, out, n_tokens, D);
  }
}